// DistillKLIdea3_46849503264830
// MI455X (gfx1250) — compile-verified
//
#include <hip/hip_runtime.h>
#include <math.h>

typedef float v2f __attribute__((ext_vector_type(2)));
typedef float v8f __attribute__((ext_vector_type(8)));

#define NEG_HUGE (-3.402823466e38f)

// ---- online softmax update for the teacher stream (tracks argmax + weighted sum S) ----
__device__ __forceinline__ void onlineT(float t, float s, int j, float invT,
                                        float& m, float& Z, float& S, int& idx) {
  const float d = t - s;
  const float e = __expf(-fabsf(t - m) * invT);  // exp(-(|t-m|)/T): scales old sums OR new term
  const bool nm = t > m;                          // strict > keeps first-occurrence argmax
  Z = nm ? __fmaf_rn(Z, e, 1.0f) : (Z + e);
  S = nm ? __fmaf_rn(S, e, d)    : __fmaf_rn(e, d, S);
  idx = nm ? j : idx;
  m = fmaxf(m, t);
}

// ---- online softmax update for the student stream (only logsumexp needed) ----
__device__ __forceinline__ void onlineS(float s, float invT, float& ms, float& Zs) {
  const float e = __expf(-fabsf(s - ms) * invT);
  const bool nm = s > ms;
  Zs = nm ? __fmaf_rn(Zs, e, 1.0f) : (Zs + e);
  ms = fmaxf(ms, s);
}

__device__ __forceinline__ void combineT(float& m, float& Z, float& S, int& idx,
                                         float om, float oZ, float oS, int oidx, float invT) {
  const float mm = fmaxf(m, om);
  const float ea = __expf((m - mm) * invT);
  const float eb = __expf((om - mm) * invT);
  Z = __fmaf_rn(Z, ea, oZ * eb);
  S = __fmaf_rn(S, ea, oS * eb);
  idx = (om > m) ? oidx : ((m > om) ? idx : min(idx, oidx));
  m = mm;
}

__device__ __forceinline__ void combineS(float& ms, float& Zs, float oms, float oZs, float invT) {
  const float mm = fmaxf(ms, oms);
  Zs = __fmaf_rn(Zs, __expf((ms - mm) * invT), oZs * __expf((oms - mm) * invT));
  ms = mm;
}

// One workgroup per row: single fused pass over y_t and y_s (bandwidth-optimal).
__global__ __launch_bounds__(256) void distill_row_kernel(
    const float* __restrict__ gys, const float* __restrict__ gyt,
    const int* __restrict__ target, float* __restrict__ rowLoss, int C) {
  constexpr float invT = 0.25f;  // 1/T, T = 4.0
  const int row = blockIdx.x;
  const int tid = threadIdx.x;
  const float* ys = gys + (size_t)row * (size_t)C;
  const float* yt = gyt + (size_t)row * (size_t)C;

  float m = NEG_HUGE, Z = 0.f, S = 0.f, ms = NEG_HUGE, Zs = 0.f;
  int idx = 0x7fffffff;

  const int C4 = C >> 2;
  const float4* ys4 = reinterpret_cast<const float4*>(ys);
  const float4* yt4 = reinterpret_cast<const float4*>(yt);
  for (int i = tid; i < C4; i += 256) {
    const float4 a = yt4[i];
    const float4 b = ys4[i];
    __builtin_prefetch(yt4 + i + 1024, 0, 0);  // global_prefetch_b8, 16 KB ahead per stream
    __builtin_prefetch(ys4 + i + 1024, 0, 0);
    const int j = i << 2;
    onlineT(a.x, b.x, j + 0, invT, m, Z, S, idx);
    onlineT(a.y, b.y, j + 1, invT, m, Z, S, idx);
    onlineT(a.z, b.z, j + 2, invT, m, Z, S, idx);
    onlineT(a.w, b.w, j + 3, invT, m, Z, S, idx);
    onlineS(b.x, invT, ms, Zs);
    onlineS(b.y, invT, ms, Zs);
    onlineS(b.z, invT, ms, Zs);
    onlineS(b.w, invT, ms, Zs);
  }
  for (int j = (C4 << 2) + tid; j < C; j += 256) {  // tail (C % 4 != 0)
    onlineT(yt[j], ys[j], j, invT, m, Z, S, idx);
    onlineS(ys[j], invT, ms, Zs);
  }

  // wave32 butterfly reduction
#pragma unroll
  for (int off = 16; off >= 1; off >>= 1) {
    const float om  = __shfl_xor(m,  off, 32);
    const float oZ  = __shfl_xor(Z,  off, 32);
    const float oS  = __shfl_xor(S,  off, 32);
    const int  oid  = __shfl_xor(idx, off, 32);
    const float oms = __shfl_xor(ms, off, 32);
    const float oZs = __shfl_xor(Zs, off, 32);
    combineT(m, Z, S, idx, om, oZ, oS, oid, invT);
    combineS(ms, Zs, oms, oZs, invT);
  }

  // 8-wave combine through LDS
  __shared__ float sm[8], sZ[8], sS[8], sms[8], sZs[8];
  __shared__ int sidx[8];
  const int wid = tid >> 5, lane = tid & 31;
  if (lane == 0) { sm[wid] = m; sZ[wid] = Z; sS[wid] = S; sms[wid] = ms; sZs[wid] = Zs; sidx[wid] = idx; }
  __syncthreads();
  if (wid == 0) {
    const bool v = lane < 8;
    m  = v ? sm[lane]  : NEG_HUGE;  Z  = v ? sZ[lane]  : 0.f;  S = v ? sS[lane] : 0.f;
    ms = v ? sms[lane] : NEG_HUGE;  Zs = v ? sZs[lane] : 0.f;  idx = v ? sidx[lane] : 0x7fffffff;
#pragma unroll
    for (int off = 4; off >= 1; off >>= 1) {
      const float om  = __shfl_xor(m,  off, 32);
      const float oZ  = __shfl_xor(Z,  off, 32);
      const float oS  = __shfl_xor(S,  off, 32);
      const int  oid  = __shfl_xor(idx, off, 32);
      const float oms = __shfl_xor(ms, off, 32);
      const float oZs = __shfl_xor(Zs, off, 32);
      combineT(m, Z, S, idx, om, oZ, oS, oid, invT);
      combineS(ms, Zs, oms, oZs, invT);
    }
    if (lane == 0) {
      // O(1) fix-up for y_t[target] += row_max (only when argmax != target)
      const int tg = target[row];
      const float yt_tg = yt[tg];
      const float ys_tg = ys[tg];
      float m2 = m, Z2 = Z, S2 = S;
      if (idx != tg) {
        const float vb = yt_tg + m;          // boosted value
        m2 = fmaxf(m, vb);
        const float sc = __expf((m - m2) * invT);
        const float eo = __expf((yt_tg - m2) * invT);
        const float en = __expf((vb - m2) * invT);
        Z2 = __fmaf_rn(Z, sc, en - eo);
        S2 = S * sc - eo * (yt_tg - ys_tg) + en * (vb - ys_tg);
      }
      // sum p_t(log p_t - log p_s) = S2/(T*Z2) - m2/T - log Z2 + ms/T + log Zs
      rowLoss[row] = S2 * invT / Z2 - m2 * invT - __logf(Z2) + ms * invT + __logf(Zs);
    }
  }
}

// Deterministic 4096 -> 1 reduction using V_WMMA_F32_16X16X4_F32 with a ones B-matrix.
// D = A*1 + C accumulates exact fp32 sums; each D element ends up the same total row-sum,
// replicated per column, so lanes 0 and 16 together hold the full sum of each 64-elem tile.
__global__ __launch_bounds__(32) void distill_final_kernel(
    const float* __restrict__ rl, float* __restrict__ out, int B, float scale) {
  const int lane = threadIdx.x;  // exactly one wave32, EXEC all ones for WMMA
  v8f d = {0.f, 0.f, 0.f, 0.f, 0.f, 0.f, 0.f, 0.f};
  v2f b;  b.x = 1.0f;  b.y = 1.0f;
  const int tiles = B >> 6;  // 64 values per WMMA (16x4 f32 A tile)
  for (int t = 0; t < tiles; ++t) {
    const float* p = rl + (t << 6) + (lane << 1);
    v2f a;  a.x = p[0];  a.y = p[1];
    d = __builtin_amdgcn_wmma_f32_16x16x4_f32(false, a, false, b, (short)0, d, false, false);
  }
  float s8 = 0.f;
#pragma unroll
  for (int v = 0; v < 8; ++v) s8 += d[v];
  const float tot = s8 + __shfl_xor(s8, 16, 32);  // lane0: M=0..7 plus lane16: M=8..15

  float tail = 0.f;  // handle B not a multiple of 64
  for (int i = (tiles << 6) + lane; i < B; i += 32) tail += rl[i];
#pragma unroll
  for (int off = 16; off >= 1; off >>= 1) tail += __shfl_xor(tail, off, 32);

  if (lane == 0) out[0] = (tot + tail) * scale;
}

extern "C" void kernel_launch(void* const* d_in, const int* in_sizes, int n_in,
                              void* d_out, int out_size, void* d_ws, size_t ws_size,
                              hipStream_t stream) {
  const float* y_s = (const float*)d_in[0];
  const float* y_t = (const float*)d_in[1];
  const int* target = (const int*)d_in[2];
  const int B = in_sizes[2];
  const int C = in_sizes[0] / B;
  float* rowLoss = (float*)d_ws;

  distill_row_kernel<<<B, 256, 0, stream>>>(y_s, y_t, target, rowLoss, C);
  distill_final_kernel<<<1, 32, 0, stream>>>(rowLoss, (float*)d_out, B, 16.0f / (float)B);
}